// Linear_77756087927635
// MI455X (gfx1250) — compile-verified
//
#include <hip/hip_runtime.h>
#include <stdint.h>

typedef float v2f __attribute__((ext_vector_type(2)));
typedef float v8f __attribute__((ext_vector_type(8)));
typedef unsigned int u32x4 __attribute__((ext_vector_type(4)));
typedef int i32x4 __attribute__((ext_vector_type(4)));
typedef int i32x8 __attribute__((ext_vector_type(8)));

#define IN_F  4096
#define OUT_F 4096
#define LDK   68   // 64 + 4 pad (16B-aligned float4 stores, conflict-free reads)
#define XK    36   // 32 + 4 pad (matches TDM pad: 4 DWORDs every 32 DWORDs)

__device__ __forceinline__ v8f wmma4(v2f a, v2f b, v8f c) {
  // D = A(16x4) * B(4x16) + C, fp32 WMMA
  return __builtin_amdgcn_wmma_f32_16x16x4_f32(false, a, false, b, (short)0, c,
                                               false, false);
}

// perm1 is an involution: within each 128-group swap chunks [32,64) <-> [64,96)
__device__ __forceinline__ int permc(int c) {
  int u = c & 96;
  return c + ((u == 32) ? 32 : (u == 64) ? -32 : 0);
}

// LDS byte offset of a __shared__ object (LDS aperture: addr[31:0] = offset)
__device__ __forceinline__ uint32_t lds_offset(const void* p) {
  return (uint32_t)(uintptr_t)p;
}

// Issue a TDM 2D tile load: rows x 32 fp32 elements, row stride = stride_elems,
// destination LDS rows padded to 36 DWORDs (pad 4 DW every 32 DW).
__device__ __forceinline__ void tdm_load_tile32(uint64_t gaddr, uint32_t laddr,
                                                int rows, int stride_elems) {
  u32x4 g0;
  g0.x = 1u;                                           // count=1 (valid), no gather
  g0.y = laddr;                                        // lds_addr (bytes)
  g0.z = (uint32_t)gaddr;                              // global_addr[31:0]
  g0.w = (uint32_t)((gaddr >> 32) & 0x01FFFFFFu) | (2u << 30);  // addr[56:32]|type=2
  i32x8 g1;
  g1[0] = (2 << 16)      // data_size = 4 bytes
        | (1 << 20)      // pad_enable
        | (4 << 22)      // pad_interval: every 32 DWORDs
        | (3 << 25);     // pad_amount: 4 DWORDs
  g1[1] = (int)((IN_F & 0xFFFF) << 16);                        // tensor_dim0 lo16
  g1[2] = (int)(((IN_F >> 16) & 0xFFFF) | ((8192 & 0xFFFF) << 16)); // d0 hi|d1 lo
  g1[3] = (int)(((8192 >> 16) & 0xFFFF) | (32 << 16));         // d1 hi | tile_dim0=32
  g1[4] = rows;                                                // tile_dim1 (tile_dim2=0)
  g1[5] = stride_elems;                                        // tensor_dim0_stride lo32
  g1[6] = 0;
  g1[7] = 0;
  i32x4 z4 = {0, 0, 0, 0};                                     // groups 2/3 unused (2D)
  i32x8 z8 = {0, 0, 0, 0, 0, 0, 0, 0};                         // trailing group unused
  __builtin_amdgcn_tensor_load_to_lds(g0, g1, z4, z4, z8, 0);
}

// ---------------------------------------------------------------------------
// K1: Cayley transform per 64x64 block: Q = (I - S)(I + S)^-1,  S = skew(R)
// ---------------------------------------------------------------------------
__global__ __launch_bounds__(256) void cayley_kernel(const float* __restrict__ R,
                                                     float* __restrict__ Q) {
  const int b = blockIdx.x;            // 0..127  (factor*64 + block)
  const float* Rb = R + b * 4096;
  float* Qb = Q + b * 4096;
  __shared__ float A[4096];            // I + S  -> reduced
  __shared__ float Ci[4096];           // -> (I+S)^-1
  __shared__ float Bm[4096];           // I - S
  __shared__ float fcol[64];
  __shared__ float pivinv;
  const int t = threadIdx.x;

  for (int idx = t; idx < 4096; idx += 256) {
    int i = idx >> 6, j = idx & 63;
    float s  = 0.5f * (Rb[i * 64 + j] - Rb[j * 64 + i]);
    float id = (i == j) ? 1.0f : 0.0f;
    A[idx]  = id + s;
    Bm[idx] = id - s;
    Ci[idx] = id;
  }
  __syncthreads();

  // Gauss-Jordan (I+S is diagonally dominant for small S; no pivoting)
  for (int k = 0; k < 64; ++k) {
    if (t == 0) pivinv = 1.0f / A[k * 64 + k];
    __syncthreads();
    if (t < 64) { A[k * 64 + t] *= pivinv; Ci[k * 64 + t] *= pivinv; }
    __syncthreads();
    if (t < 64) fcol[t] = (t == k) ? 0.0f : A[t * 64 + k];
    __syncthreads();
    for (int idx = t; idx < 4096; idx += 256) {
      int i = idx >> 6, j = idx & 63;
      if (i != k) {
        float f = fcol[i];
        A[idx]  -= f * A[k * 64 + j];
        Ci[idx] -= f * Ci[k * 64 + j];
      }
    }
    __syncthreads();
  }

  // Q = (I - S) @ (I+S)^-1
  for (int idx = t; idx < 4096; idx += 256) {
    int i = idx >> 6, j = idx & 63;
    float acc = 0.0f;
    for (int k = 0; k < 64; ++k) acc += Bm[i * 64 + k] * Ci[k * 64 + j];
    Qb[idx] = acc;
  }
}

// ---------------------------------------------------------------------------
// K2: W1 = weight @ m0^T   (m0 = blockdiag(Q0), perm0 == identity)
//     W1[o, cb*64+j] = sum_k weight[o, cb*64+k] * Q0[cb][j,k]
// ---------------------------------------------------------------------------
__global__ __launch_bounds__(256) void factor0_kernel(const float* __restrict__ W,
                                                      const float* __restrict__ Q,
                                                      float* __restrict__ W1) {
  const int rowgrp = blockIdx.x;       // 64 weight rows
  const int cb     = blockIdx.y;       // 64-column block
  __shared__ float As[64 * LDK];
  __shared__ float Bs[64 * LDK];
  const int t = threadIdx.x;
  const float* Qb = Q + cb * 4096;     // factor 0

  for (int i = 0; i < 4; ++i) {
    int lin = (t + i * 256) * 4;       // 4096 floats each tile
    int r = lin >> 6, c = lin & 63;
    float4 va = *reinterpret_cast<const float4*>(
        &W[(size_t)(rowgrp * 64 + r) * IN_F + cb * 64 + c]);
    *reinterpret_cast<float4*>(&As[r * LDK + c]) = va;
    float4 vb = *reinterpret_cast<const float4*>(&Qb[lin]);
    *reinterpret_cast<float4*>(&Bs[r * LDK + c]) = vb;
  }
  __syncthreads();

  const int wid = t >> 5, lane = t & 31;
  const int lrow = lane & 15, lhi = lane >> 4;
  const int tr = wid >> 1, tc0 = (wid & 1) * 2;

  v8f acc0 = {}, acc1 = {};
  for (int kk = 0; kk < 64; kk += 4) {
    v2f a  = *reinterpret_cast<const v2f*>(&As[(tr * 16 + lrow) * LDK + kk + 2 * lhi]);
    v2f b0 = *reinterpret_cast<const v2f*>(&Bs[((tc0 + 0) * 16 + lrow) * LDK + kk + 2 * lhi]);
    v2f b1 = *reinterpret_cast<const v2f*>(&Bs[((tc0 + 1) * 16 + lrow) * LDK + kk + 2 * lhi]);
    acc0 = wmma4(a, b0, acc0);
    acc1 = wmma4(a, b1, acc1);
  }

  const int j0 = cb * 64 + (tc0 + 0) * 16 + lrow;
  const int j1 = cb * 64 + (tc0 + 1) * 16 + lrow;
  for (int r = 0; r < 8; ++r) {
    int o = rowgrp * 64 + tr * 16 + r + 8 * lhi;
    W1[(size_t)o * IN_F + j0] = acc0[r];
    W1[(size_t)o * IN_F + j1] = acc1[r];
  }
}

// ---------------------------------------------------------------------------
// K3: W_eff = (W1 @ m1^T) * s   with m1 = P1 blockdiag(Q1) P1^T
//     gather columns by perm on load, scatter by perm + scale on store
// ---------------------------------------------------------------------------
__global__ __launch_bounds__(256) void factor1_kernel(const float* __restrict__ W1,
                                                      const float* __restrict__ Q,
                                                      const float* __restrict__ s,
                                                      float* __restrict__ We) {
  const int rowgrp = blockIdx.x;
  const int blk    = blockIdx.y;
  __shared__ float As[64 * LDK];
  __shared__ float Bs[64 * LDK];
  const int t = threadIdx.x;
  const float* Qb = Q + (size_t)(64 + blk) * 4096;   // factor 1

  for (int i = 0; i < 4; ++i) {
    int lin = (t + i * 256) * 4;
    int r = lin >> 6, c = lin & 63;
    int srccol = permc(blk * 64 + c);   // 32-chunk moves keep float4 contiguous
    float4 va = *reinterpret_cast<const float4*>(
        &W1[(size_t)(rowgrp * 64 + r) * IN_F + srccol]);
    *reinterpret_cast<float4*>(&As[r * LDK + c]) = va;
    float4 vb = *reinterpret_cast<const float4*>(&Qb[lin]);
    *reinterpret_cast<float4*>(&Bs[r * LDK + c]) = vb;
  }
  __syncthreads();

  const int wid = t >> 5, lane = t & 31;
  const int lrow = lane & 15, lhi = lane >> 4;
  const int tr = wid >> 1, tc0 = (wid & 1) * 2;

  v8f acc0 = {}, acc1 = {};
  for (int kk = 0; kk < 64; kk += 4) {
    v2f a  = *reinterpret_cast<const v2f*>(&As[(tr * 16 + lrow) * LDK + kk + 2 * lhi]);
    v2f b0 = *reinterpret_cast<const v2f*>(&Bs[((tc0 + 0) * 16 + lrow) * LDK + kk + 2 * lhi]);
    v2f b1 = *reinterpret_cast<const v2f*>(&Bs[((tc0 + 1) * 16 + lrow) * LDK + kk + 2 * lhi]);
    acc0 = wmma4(a, b0, acc0);
    acc1 = wmma4(a, b1, acc1);
  }

  const int a0 = permc(blk * 64 + (tc0 + 0) * 16 + lrow);
  const int a1 = permc(blk * 64 + (tc0 + 1) * 16 + lrow);
  for (int r = 0; r < 8; ++r) {
    int o = rowgrp * 64 + tr * 16 + r + 8 * lhi;
    float sc = s[o];
    We[(size_t)o * IN_F + a0] = acc0[r] * sc;
    We[(size_t)o * IN_F + a1] = acc1[r] * sc;
  }
}

// ---------------------------------------------------------------------------
// K4: out[m,n] = sum_k x[m,k] * W_eff[n,k] + bias[n]   (8192 x 4096 x 4096)
//     128x64 tile per WG; TDM double-buffered LDS staging; N-fastest grid so
//     W_eff (67 MB) stays L2-resident while x streams once from HBM.
// ---------------------------------------------------------------------------
__global__ __launch_bounds__(256) void gemm_kernel(const float* __restrict__ x,
                                                   const float* __restrict__ We,
                                                   const float* __restrict__ bias,
                                                   float* __restrict__ out) {
  const int nt = blockIdx.x;           // 64 col tiles of 64  (fastest -> L2 reuse)
  const int mt = blockIdx.y;           // 64 row tiles of 128
  __shared__ float Xs[2 * 128 * XK];
  __shared__ float Ws[2 * 64 * XK];
  const int t = threadIdx.x;
  const int wid = t >> 5, lane = t & 31;
  const int lrow = lane & 15, lhi = lane >> 4;
  const int wr = wid & 3, wc = wid >> 2;
  const int m0 = mt * 128, n0 = nt * 64;

  const uint64_t xbase = (uint64_t)(uintptr_t)x  + ((uint64_t)m0 * IN_F) * 4;
  const uint64_t wbase = (uint64_t)(uintptr_t)We + ((uint64_t)n0 * IN_F) * 4;
  const uint32_t xlds = lds_offset(&Xs[0]);
  const uint32_t wlds = lds_offset(&Ws[0]);

  v8f acc[2][2] = {{{0.f}, {0.f}}, {{0.f}, {0.f}}};

  // prologue: DMA chunk 0 into buffer 0 (EXEC-independent, SGPR descriptor)
  if (t == 0) {
    tdm_load_tile32(xbase, xlds, 128, IN_F);
    tdm_load_tile32(wbase, wlds, 64, IN_F);
  }

  const int nchunks = IN_F / 32;
  for (int i = 0; i < nchunks; ++i) {
    const int cur = i & 1, nxt = cur ^ 1;
    if (t == 0) {
      if (i + 1 < nchunks) {
        // overlap: DMA chunk i+1 into the other buffer, then wait for chunk i
        uint64_t off = (uint64_t)(i + 1) * 32 * 4;
        tdm_load_tile32(xbase + off, xlds + (uint32_t)(nxt * 128 * XK) * 4, 128, IN_F);
        tdm_load_tile32(wbase + off, wlds + (uint32_t)(nxt * 64 * XK) * 4, 64, IN_F);
        __builtin_amdgcn_s_wait_tensorcnt(2);  // in-order: chunk i's 2 tiles landed
      } else {
        __builtin_amdgcn_s_wait_tensorcnt(0);
      }
    }
    __syncthreads();

    const float* Xb = &Xs[cur * 128 * XK];
    const float* Wb = &Ws[cur * 64 * XK];
    for (int kk = 0; kk < 32; kk += 4) {
      v2f a0 = *reinterpret_cast<const v2f*>(&Xb[(wr * 32 +      lrow) * XK + kk + 2 * lhi]);
      v2f a1 = *reinterpret_cast<const v2f*>(&Xb[(wr * 32 + 16 + lrow) * XK + kk + 2 * lhi]);
      v2f b0 = *reinterpret_cast<const v2f*>(&Wb[(wc * 32 +      lrow) * XK + kk + 2 * lhi]);
      v2f b1 = *reinterpret_cast<const v2f*>(&Wb[(wc * 32 + 16 + lrow) * XK + kk + 2 * lhi]);
      acc[0][0] = wmma4(a0, b0, acc[0][0]);
      acc[0][1] = wmma4(a0, b1, acc[0][1]);
      acc[1][0] = wmma4(a1, b0, acc[1][0]);
      acc[1][1] = wmma4(a1, b1, acc[1][1]);
    }
    __syncthreads();  // all waves done with buf cur before it is DMA-overwritten
  }

  for (int j = 0; j < 2; ++j) {
    int n = n0 + wc * 32 + j * 16 + lrow;
    float bv = bias[n];
    for (int i = 0; i < 2; ++i) {
      for (int r = 0; r < 8; ++r) {
        int m = m0 + wr * 32 + i * 16 + r + 8 * lhi;
        out[(size_t)m * OUT_F + n] = acc[i][j][r] + bv;
      }
    }
  }
}

// ---------------------------------------------------------------------------
extern "C" void kernel_launch(void* const* d_in, const int* in_sizes, int n_in,
                              void* d_out, int out_size, void* d_ws, size_t ws_size,
                              hipStream_t stream) {
  const float* x      = (const float*)d_in[0];
  const float* weight = (const float*)d_in[1];
  const float* bias   = (const float*)d_in[2];
  const float* boft_R = (const float*)d_in[3];
  const float* boft_s = (const float*)d_in[4];
  float* out = (float*)d_out;

  float* Q  = (float*)d_ws;                        // 2*64*64*64 floats (2 MB)
  float* W1 = Q + (size_t)2 * 64 * 64 * 64;        // 4096*4096 floats (64 MB)
  float* We = W1 + (size_t)OUT_F * IN_F;           // 4096*4096 floats (64 MB)

  cayley_kernel <<<128, 256, 0, stream>>>(boft_R, Q);
  factor0_kernel<<<dim3(64, 64), 256, 0, stream>>>(weight, Q, W1);
  factor1_kernel<<<dim3(64, 64), 256, 0, stream>>>(W1, Q, boft_s, We);
  gemm_kernel   <<<dim3(64, 64), 256, 0, stream>>>(x, We, bias, out);
}